// ToRGB_10393820857038
// MI455X (gfx1250) — compile-verified
//
#include <hip/hip_runtime.h>
#include <hip/hip_bf16.h>
#include <math.h>

typedef __attribute__((ext_vector_type(2))) float v2f;
typedef __attribute__((ext_vector_type(8))) float v8f;

#define IN_CH      512
#define HW         128
#define CH_STRIDE  (HW * HW)      // 16384 floats between channels of x / out
#define SKIP_HW    64
#define SKIP_CH    (SKIP_HW * SKIP_HW) // 4096
#define SLOPE      0.2f
#define ACT_SCALE  1.4142135623730951f
#define WSCALE     0.04419417382415922f  // 1/sqrt(512)

// One wave handles 16 consecutive output pixels (same b,h; w = w0..w0+15).
// GEMM via V_WMMA_F32_16X16X4_F32: A = zero-padded scaled weights [16 x 4],
// B = x slice [4 channels x 16 pixels], accumulate 128 K-slices.
__global__ __launch_bounds__(256) void torgb_kernel(
    const float* __restrict__ x,     // [8,512,128,128]
    const float* __restrict__ skip,  // [8,3,64,64]
    const float* __restrict__ cw,    // [3,512]
    const float* __restrict__ flrb,  // [3]
    const float* __restrict__ rgbb,  // [3]
    float* __restrict__ out)         // [8,3,128,128]
{
    __shared__ float wlds[16 * IN_CH]; // 32 KB: rows 0..2 = scaled weights, 3..15 = 0

    const int tid = threadIdx.x;
    for (int idx = tid; idx < 16 * IN_CH; idx += 256) {
        const int m = idx >> 9;        // row (out channel)
        const int c = idx & (IN_CH - 1);
        wlds[idx] = (m < 3) ? cw[m * IN_CH + c] * WSCALE : 0.0f;
    }
    __syncthreads();

    const int lane = tid & 31;
    const int wave = tid >> 5;
    const int g    = blockIdx.x * 8 + wave;  // pixel-group id, 16 pixels each
    const int s0   = g << 4;
    const int b    = s0 >> 14;               // / (128*128)
    const int rem  = s0 & (CH_STRIDE - 1);
    const int h    = rem >> 7;
    const int w0   = rem & (HW - 1);         // multiple of 16
    const int n    = lane & 15;              // matrix row/col index for this lane
    const int koff = (lane >> 4) << 1;       // K offset: lanes 0-15 -> K0/K1, 16-31 -> K2/K3

    // Per-lane base into x: pixel (b, h, w0+n), channel 0
    const float* xbase = x + (size_t)b * IN_CH * CH_STRIDE + (size_t)h * HW + (w0 + n);

    v8f acc = {};
    #pragma unroll 4
    for (int cb = 0; cb < IN_CH; cb += 4) {
        // A: weights, ds_load_b64 (8B-aligned: cb%4==0, koff even)
        v2f a = *reinterpret_cast<const v2f*>(&wlds[(n << 9) + cb + koff]);
        // B: x, two coalesced 32b loads (channel stride 64KB apart)
        const float* p = xbase + (size_t)(cb + koff) * CH_STRIDE;
        __builtin_prefetch(p + (size_t)8 * CH_STRIDE, 0, 0); // global_prefetch_b8, ~2 iters ahead
        v2f bv;
        bv.x = p[0];
        bv.y = p[CH_STRIDE];
        acc = __builtin_amdgcn_wmma_f32_16x16x4_f32(
            /*neg_a=*/false, a, /*neg_b=*/false, bv,
            /*c_mod=*/(short)0, acc, /*reuse_a=*/false, /*reuse_b=*/false);
    }

    // Epilogue: lanes 0-15 hold rows 0..2 (R,G,B) for pixel (b,h,w0+n)
    if (lane < 16) {
        const int wp = w0 + n;

        // 2x upsample taps (separable [1,3,3,1]/4 per axis after up-scaling)
        int hi0, hi1, wi0, wi1;
        float hf0, hf1, wf0, wf1;
        if (h & 1) { hi0 = (h - 1) >> 1; hf0 = 0.75f; hi1 = (h + 1) >> 1; hf1 = 0.25f; }
        else       { hi0 = (h >> 1) - 1; hf0 = 0.25f; hi1 =  h >> 1;      hf1 = 0.75f; }
        if (wp & 1){ wi0 = (wp - 1) >> 1; wf0 = 0.75f; wi1 = (wp + 1) >> 1; wf1 = 0.25f; }
        else       { wi0 = (wp >> 1) - 1; wf0 = 0.25f; wi1 =  wp >> 1;      wf1 = 0.75f; }
        const bool h0ok = (hi0 >= 0) && (hi0 < SKIP_HW);
        const bool h1ok = (hi1 >= 0) && (hi1 < SKIP_HW);
        const bool w0ok = (wi0 >= 0) && (wi0 < SKIP_HW);
        const bool w1ok = (wi1 >= 0) && (wi1 < SKIP_HW);

        const float* sb = skip + (size_t)b * 3 * SKIP_CH;
        float* ob = out + (size_t)b * 3 * CH_STRIDE + (size_t)h * HW + wp;

        #pragma unroll
        for (int o = 0; o < 3; ++o) {
            float z = acc[o] + flrb[o];
            z = (z >= 0.0f ? z : SLOPE * z) * ACT_SCALE;
            z += rgbb[o];

            const float* sc = sb + (size_t)o * SKIP_CH;
            float up = 0.0f;
            if (h0ok) {
                if (w0ok) up += hf0 * wf0 * sc[hi0 * SKIP_HW + wi0];
                if (w1ok) up += hf0 * wf1 * sc[hi0 * SKIP_HW + wi1];
            }
            if (h1ok) {
                if (w0ok) up += hf1 * wf0 * sc[hi1 * SKIP_HW + wi0];
                if (w1ok) up += hf1 * wf1 * sc[hi1 * SKIP_HW + wi1];
            }
            ob[(size_t)o * CH_STRIDE] = z + up;
        }
    }
}

extern "C" void kernel_launch(void* const* d_in, const int* in_sizes, int n_in,
                              void* d_out, int out_size, void* d_ws, size_t ws_size,
                              hipStream_t stream) {
    (void)in_sizes; (void)n_in; (void)d_ws; (void)ws_size; (void)out_size;
    const float* x    = (const float*)d_in[0];
    const float* skip = (const float*)d_in[1];
    const float* cw   = (const float*)d_in[2];
    const float* fb   = (const float*)d_in[3];
    const float* rb   = (const float*)d_in[4];
    float* out = (float*)d_out;

    // 131072 pixels / 16 per wave / 8 waves per block = 1024 blocks
    dim3 grid(1024), block(256);
    hipLaunchKernelGGL(torgb_kernel, grid, block, 0, stream,
                       x, skip, cw, fb, rb, out);
}